// Selection_11914239279107
// MI455X (gfx1250) — compile-verified
//
#include <hip/hip_runtime.h>

#define N_EXPERTS 8
#define D_IN 1024
#define D_OUT 1024
#define BATCH 4096

#define MB 64          // rows (samples) per block tile
#define NB 64          // output features per block tile
#define KB 32          // K-step per LDS stage
#define KPAD 36        // padded LDS row stride (floats): 144B, 16B aligned, conflict-free
#define NKB (D_IN / KB)

typedef __attribute__((ext_vector_type(2))) float v2f;
typedef __attribute__((ext_vector_type(8))) float v8f;

__device__ __forceinline__ v8f wmma_f32_k4(v2f a, v2f b, v8f c) {
  // D = A(16x4 f32) * B(4x16 f32) + C(16x16 f32), exact f32
  return __builtin_amdgcn_wmma_f32_16x16x4_f32(false, a, false, b, (short)0, c,
                                               false, false);
}

// Async copy global -> LDS (CDNA5, tracked by ASYNCcnt; no VGPR staging).
__device__ __forceinline__ void async_load_b128(unsigned lds_off,
                                                const float* gaddr) {
  asm volatile("global_load_async_to_lds_b128 %0, %1, off"
               :: "v"(lds_off), "v"(gaddr)
               : "memory");
}

__device__ __forceinline__ unsigned lds_offset(const void* p) {
  // Low 32 bits of the generic address of a __shared__ object are the LDS
  // byte offset (ISA 10.2: LDS aperture).
  return (unsigned)(unsigned long long)p;
}

// ---------------------------------------------------------------------------
// Kernel 1: bucket samples by expert. ws layout: counts[8] | offsets[8] | perm[4096]
// ---------------------------------------------------------------------------
__global__ void build_perm_kernel(const void* __restrict__ actions_raw, int act64,
                                  int* __restrict__ counts, int* __restrict__ offsets,
                                  int* __restrict__ perm, int batch) {
  __shared__ int s_cnt[N_EXPERTS];
  __shared__ int s_off[N_EXPERTS];
  __shared__ int s_cur[N_EXPERTS];
  const int tid = threadIdx.x;
  if (tid < N_EXPERTS) { s_cnt[tid] = 0; s_cur[tid] = 0; }
  __syncthreads();
  const int* a32 = (const int*)actions_raw;
  const long long* a64 = (const long long*)actions_raw;
  for (int i = tid; i < batch; i += blockDim.x) {
    int e = (act64 ? (int)a64[i] : a32[i]) & (N_EXPERTS - 1);
    atomicAdd(&s_cnt[e], 1);
  }
  __syncthreads();
  if (tid == 0) {
    int run = 0;
    for (int e = 0; e < N_EXPERTS; ++e) { s_off[e] = run; run += s_cnt[e]; }
  }
  __syncthreads();
  for (int i = tid; i < batch; i += blockDim.x) {
    int e = (act64 ? (int)a64[i] : a32[i]) & (N_EXPERTS - 1);
    int p = atomicAdd(&s_cur[e], 1);
    perm[s_off[e] + p] = i;
  }
  if (tid < N_EXPERTS) { counts[tid] = s_cnt[tid]; offsets[tid] = s_off[tid]; }
}

// ---------------------------------------------------------------------------
// Kernel 2: grouped GEMM, one 64x64 output tile per block for expert blockIdx.z
// Double-buffered LDS tiles fed by async global->LDS DMA, f32 WMMA compute.
// ---------------------------------------------------------------------------
__global__ __launch_bounds__(256) void moe_gemm_kernel(
    const float* __restrict__ xs, const float* __restrict__ W,
    const float* __restrict__ bias, const int* __restrict__ counts,
    const int* __restrict__ offsets, const int* __restrict__ perm,
    float* __restrict__ out) {
  const int e = blockIdx.z;
  const int cnt = counts[e];
  const int m0 = blockIdx.y * MB;
  if (m0 >= cnt) return;   // wave-uniform early exit for unused m-tiles
  const int n0 = blockIdx.x * NB;
  const int off = offsets[e];

  __shared__ __align__(16) float sA[2][MB][KPAD];
  __shared__ __align__(16) float sB[2][NB][KPAD];
  __shared__ int s_row[MB];

  const int tid = threadIdx.x;
  if (tid < MB) {
    int g = m0 + tid;
    s_row[tid] = (g < cnt) ? perm[off + g] : perm[off];  // clamp to valid sample
  }
  __syncthreads();

  const int lane = tid & 31;
  const int wave = tid >> 5;
  const int wn = wave & 3;     // 4 waves across N (16 cols each)
  const int wm = wave >> 2;    // 2 waves across M (32 rows each)
  const int half = lane >> 4;  // K-pair select per ISA 16x4 A layout
  const int lm = lane & 15;    // row/col within a 16x16 tile

  const float* Wg = W + ((size_t)e * D_OUT + n0) * D_IN;

  // Per-thread tile-slice coordinates: 2 x b128 for A and B each.
  const int r0 = tid >> 3;                 // 0..31
  const int r1 = (tid + 256) >> 3;         // 32..63
  const int c4 = (tid & 7) * 4;            // 0,4,...,28

  auto asyncLoadTile = [&](int kb, int buf) {
    const int k0 = kb * KB;
    async_load_b128(lds_offset(&sA[buf][r0][c4]),
                    xs + (size_t)s_row[r0] * D_IN + k0 + c4);
    async_load_b128(lds_offset(&sB[buf][r0][c4]),
                    Wg + (size_t)r0 * D_IN + k0 + c4);
    async_load_b128(lds_offset(&sA[buf][r1][c4]),
                    xs + (size_t)s_row[r1] * D_IN + k0 + c4);
    async_load_b128(lds_offset(&sB[buf][r1][c4]),
                    Wg + (size_t)r1 * D_IN + k0 + c4);
  };

  v8f acc0 = {};  // rows wm*32 + 0..15
  v8f acc1 = {};  // rows wm*32 + 16..31

  asyncLoadTile(0, 0);
  asm volatile("s_wait_asynccnt 0x0" ::: "memory");
  __syncthreads();

  for (int kb = 0; kb < NKB; ++kb) {
    const int cur = kb & 1;
    const bool hasNext = (kb + 1) < NKB;
    if (hasNext) asyncLoadTile(kb + 1, cur ^ 1);  // DMA overlaps WMMA compute
#pragma unroll
    for (int kk = 0; kk < KB; kk += 4) {
      const int kq = kk + 2 * half;
      v2f bf  = *(const v2f*)&sB[cur][wn * 16 + lm][kq];
      v2f af0 = *(const v2f*)&sA[cur][wm * 32 + lm][kq];
      v2f af1 = *(const v2f*)&sA[cur][wm * 32 + 16 + lm][kq];
      acc0 = wmma_f32_k4(af0, bf, acc0);
      acc1 = wmma_f32_k4(af1, bf, acc1);
    }
    if (hasNext) {
      asm volatile("s_wait_asynccnt 0x0" ::: "memory");
      __syncthreads();
    }
  }

  // Epilogue: bias + scatter back to original sample rows (EXEC divergence OK here)
  const int n = n0 + wn * 16 + lm;
  const float bv = bias[(size_t)e * D_OUT + n];
#pragma unroll
  for (int j = 0; j < 8; ++j) {
    // C/D layout: VGPR j holds M=j (lanes 0-15) and M=j+8 (lanes 16-31)
    int loc0 = wm * 32 + j + 8 * half;
    if (m0 + loc0 < cnt)
      out[(size_t)s_row[loc0] * D_OUT + n] = acc0[j] + bv;
    int loc1 = wm * 32 + 16 + j + 8 * half;
    if (m0 + loc1 < cnt)
      out[(size_t)s_row[loc1] * D_OUT + n] = acc1[j] + bv;
  }
}

// ---------------------------------------------------------------------------
// Kernel 3: pass-through of mxs and actions into concatenated output tail
// ---------------------------------------------------------------------------
__global__ void copy_tail_kernel(const int* __restrict__ mxs,
                                 const void* __restrict__ actions_raw, int act64,
                                 float* __restrict__ out, int batch) {
  const int i = blockIdx.x * blockDim.x + threadIdx.x;
  int* oi = (int*)(out + (size_t)BATCH * D_OUT);
  if (i < batch) oi[i] = mxs[i];
  const int* aw = (const int*)actions_raw;
  const int nw = act64 ? batch * 2 : batch;  // raw 4-byte words of actions
  if (i < nw) oi[batch + i] = aw[i];
}

extern "C" void kernel_launch(void* const* d_in, const int* in_sizes, int n_in,
                              void* d_out, int out_size, void* d_ws, size_t ws_size,
                              hipStream_t stream) {
  const float* xs = (const float*)d_in[0];
  const int* mxs = (const int*)d_in[1];
  const void* actions = d_in[2];
  const float* W = (const float*)d_in[3];
  const float* bias = (const float*)d_in[4];
  float* out = (float*)d_out;
  int* ws = (int*)d_ws;

  // Detect actions dtype (int32 vs int64) from the flat output element count.
  const int base = BATCH * D_OUT + BATCH;  // ys + mxs
  const int act64 = (out_size >= base + 2 * BATCH) ? 1 : 0;

  int* counts = ws;
  int* offsets = ws + N_EXPERTS;
  int* perm = ws + 2 * N_EXPERTS;

  build_perm_kernel<<<1, 512, 0, stream>>>(actions, act64, counts, offsets, perm,
                                           BATCH);

  dim3 grid(D_OUT / NB, (BATCH + MB - 1) / MB, N_EXPERTS);
  moe_gemm_kernel<<<grid, 256, 0, stream>>>(xs, W, bias, counts, offsets, perm,
                                            out);

  copy_tail_kernel<<<(2 * BATCH + 255) / 256, 256, 0, stream>>>(mxs, actions,
                                                                act64, out, BATCH);
}